// GraphClassifier_17695265259720
// MI455X (gfx1250) — compile-verified
//
#include <hip/hip_runtime.h>

typedef __attribute__((ext_vector_type(2))) float v2f;
typedef __attribute__((ext_vector_type(8))) float v8f;

#define HID 128
#define AS_STRIDE 132  // 128 + 4 pad: conflict-free A-fragment reads

// ---------------- degree / norm ----------------
__global__ __launch_bounds__(256) void k_init_deg(float* deg, int N) {
  int i = blockIdx.x * 256 + threadIdx.x;
  if (i < N) deg[i] = 1.0f;  // self loop contributes 1 to degree
}

__global__ __launch_bounds__(256) void k_count_deg(const int* __restrict__ row, float* deg, int E) {
  int e = blockIdx.x * 256 + threadIdx.x;
  if (e < E) atomicAdd(&deg[row[e]], 1.0f);
}

__global__ __launch_bounds__(256) void k_dis(float* deg_dis, int N) {
  int i = blockIdx.x * 256 + threadIdx.x;
  if (i < N) deg_dis[i] = rsqrtf(deg_dis[i] + 1e-12f);
}

__global__ __launch_bounds__(256) void k_norm(const int* __restrict__ row, const int* __restrict__ col,
                                              const float* __restrict__ dis, float* __restrict__ norm,
                                              int E) {
  int e = blockIdx.x * 256 + threadIdx.x;
  if (e < E) norm[e] = dis[row[e]] * dis[col[e]];
}

// ---------------- dense h = x@W + b via V_WMMA_F32_16X16X4_F32 ----------------
// 256 threads = 8 waves; block owns 16 rows, wave w owns cols [16w, 16w+16).
__global__ __launch_bounds__(256) void k_gemm128(const float* __restrict__ x,
                                                 const float* __restrict__ W,
                                                 const float* __restrict__ bias,
                                                 float* __restrict__ h,
                                                 int N, int relu_in) {
  __shared__ float As[16 * AS_STRIDE];
  const int tid = threadIdx.x;
  const int row0 = blockIdx.x * 16;

  // stage 16x128 A tile (coalesced), fold input ReLU here (layers 2/3)
  for (int i = tid; i < 16 * HID; i += 256) {
    int r = i >> 7, c = i & 127;
    int gr = row0 + r;
    float v = (gr < N) ? x[(size_t)gr * HID + c] : 0.0f;
    if (relu_in) v = fmaxf(v, 0.0f);
    As[r * AS_STRIDE + c] = v;
  }
  __syncthreads();

  const int lane  = tid & 31;
  const int wave  = tid >> 5;
  const int l16   = lane & 15;
  const int khalf = lane >> 4;   // 0 or 1 -> K offset 0 or 2
  const int n0    = wave * 16;

  v8f acc = {};
  const float* As_row = &As[l16 * AS_STRIDE];
  for (int k0 = 0; k0 < HID; k0 += 4) {
    const int ka = k0 + 2 * khalf;
    v2f a, b;
    a.x = As_row[ka];
    a.y = As_row[ka + 1];
    b.x = W[ka * HID + n0 + l16];         // W is 64KB -> resident in L2/WGP$
    b.y = W[(ka + 1) * HID + n0 + l16];
    acc = __builtin_amdgcn_wmma_f32_16x16x4_f32(false, a, false, b,
                                                (short)0, acc, false, false);
  }

  const float bv = bias[n0 + l16];
  for (int r = 0; r < 8; ++r) {
    int gm = row0 + r + 8 * khalf;        // C/D layout: M = r + 8*(lane>>4)
    if (gm < N) h[(size_t)gm * HID + n0 + l16] = acc[r] + bv;
  }
}

// ---------------- aggregation: out[i] = h[i]*dis[i]^2 ; out[row] += h[col]*norm ----------------
__global__ __launch_bounds__(256) void k_agg_self(const float* __restrict__ h,
                                                  const float* __restrict__ dis,
                                                  float* __restrict__ out, int total) {
  int i = blockIdx.x * 256 + threadIdx.x;
  if (i < total) {
    float d = dis[i >> 7];
    out[i] = h[i] * d * d;
  }
}

// one wave per edge: 32 lanes x float4 = full 128-float row, coalesced
__global__ __launch_bounds__(512) void k_agg_edges(const float* __restrict__ h,
                                                   const int* __restrict__ row,
                                                   const int* __restrict__ col,
                                                   const float* __restrict__ norm,
                                                   float* __restrict__ out, int E) {
  int gt   = blockIdx.x * 512 + threadIdx.x;
  int e    = gt >> 5;
  int lane = gt & 31;
  if (e >= E) return;
  int   r  = row[e];
  int   c  = col[e];
  float nv = norm[e];
  const float4* hp = (const float4*)(h + (size_t)c * HID);
  float4 v = hp[lane];
  float* op = out + (size_t)r * HID + lane * 4;
  atomicAdd(op + 0, v.x * nv);
  atomicAdd(op + 1, v.y * nv);
  atomicAdd(op + 2, v.z * nv);
  atomicAdd(op + 3, v.w * nv);
}

// ---------------- pooling (mean + max per graph) ----------------
__global__ __launch_bounds__(256) void k_pool_zero(float* s, float* mx, float* cnt, int G) {
  int i = blockIdx.x * 256 + threadIdx.x;
  if (i < G * HID) { s[i] = 0.0f; mx[i] = 0.0f; }
  if (i < G) cnt[i] = 0.0f;
}

__global__ __launch_bounds__(256) void k_pool_accum(const float* __restrict__ x,
                                                    const int* __restrict__ batch,
                                                    float* s, unsigned int* mx, float* cnt,
                                                    int total) {
  int i = blockIdx.x * 256 + threadIdx.x;
  if (i >= total) return;
  int node = i >> 7, f = i & 127;
  int g = batch[node];
  float v = fmaxf(x[i], 0.0f);            // ReLU of final GCN layer
  atomicAdd(&s[g * HID + f], v);
  atomicMax(&mx[g * HID + f], __float_as_uint(v));  // valid: v >= 0
  if (f == 0) atomicAdd(&cnt[g], 1.0f);
}

__global__ __launch_bounds__(256) void k_pool_final(const float* __restrict__ s,
                                                    const float* __restrict__ mxf,
                                                    const float* __restrict__ cnt,
                                                    float* __restrict__ gbuf, int G) {
  int i = blockIdx.x * 256 + threadIdx.x;
  if (i >= G * 2 * HID) return;
  int g = i >> 8, f = i & 255;
  if (f < HID) gbuf[i] = s[g * HID + f] / (cnt[g] + 1e-12f);
  else         gbuf[i] = mxf[g * HID + (f - HID)];
}

// ---------------- tiny MLP head ----------------
__global__ __launch_bounds__(256) void k_mlp1(const float* __restrict__ g,
                                              const float* __restrict__ Wm1,
                                              const float* __restrict__ bm1,
                                              float* __restrict__ h, int G) {
  int i = blockIdx.x * 256 + threadIdx.x;
  if (i >= G * HID) return;
  int gi = i >> 7, j = i & 127;
  const float* gr = g + (size_t)gi * 2 * HID;
  float acc = bm1[j];
  for (int k = 0; k < 2 * HID; ++k) acc += gr[k] * Wm1[k * HID + j];
  h[i] = fmaxf(acc, 0.0f);
}

__global__ __launch_bounds__(64) void k_mlp2(const float* __restrict__ h,
                                             const float* __restrict__ Wm2,
                                             const float* __restrict__ bm2,
                                             float* __restrict__ out, int G) {
  int i = blockIdx.x * 64 + threadIdx.x;
  if (i >= G * 10) return;
  int gi = i / 10, j = i % 10;
  const float* hr = h + (size_t)gi * HID;
  float acc = bm2[j];
  for (int k = 0; k < HID; ++k) acc += hr[k] * Wm2[k * 10 + j];
  out[i] = acc;
}

// ---------------- host orchestration ----------------
extern "C" void kernel_launch(void* const* d_in, const int* in_sizes, int n_in,
                              void* d_out, int out_size, void* d_ws, size_t ws_size,
                              hipStream_t stream) {
  const float* X     = (const float*)d_in[0];
  const int*   EI    = (const int*)d_in[1];
  const int*   batch = (const int*)d_in[2];
  // d_in[3] = num_graphs scalar (on device); G derived from out_size instead
  const float* W1  = (const float*)d_in[4];
  const float* b1  = (const float*)d_in[5];
  const float* W2  = (const float*)d_in[6];
  const float* b2  = (const float*)d_in[7];
  const float* W3  = (const float*)d_in[8];
  const float* b3  = (const float*)d_in[9];
  const float* Wm1 = (const float*)d_in[10];
  const float* bm1 = (const float*)d_in[11];
  const float* Wm2 = (const float*)d_in[12];
  const float* bm2 = (const float*)d_in[13];
  float* out = (float*)d_out;

  const int N = in_sizes[0] / HID;   // 100000
  const int E = in_sizes[1] / 2;     // 1600000
  const int G = out_size / 10;       // 128

  const int* row = EI;
  const int* col = EI + E;

  // carve workspace (256B aligned slices)
  char* wp = (char*)d_ws;
  auto carve = [&](size_t nfloats) {
    float* p = (float*)wp;
    wp += ((nfloats * sizeof(float) + 255) / 256) * 256;
    return p;
  };
  float* h0    = carve((size_t)N * HID);
  float* h1    = carve((size_t)N * HID);
  float* dis   = carve((size_t)N);        // deg then dis in-place
  float* normv = carve((size_t)E);
  float* s     = carve((size_t)G * HID);
  float* mx    = carve((size_t)G * HID);
  float* cnt   = carve((size_t)G);
  float* gbuf  = carve((size_t)G * 2 * HID);
  float* hm    = carve((size_t)G * HID);
  (void)ws_size;

  const int NH = N * HID;
  const unsigned gN   = (N + 255) / 256;
  const unsigned gE   = (E + 255) / 256;
  const unsigned gNH  = (NH + 255) / 256;
  const unsigned gTil = (N + 15) / 16;
  const unsigned gEdg = (E + 15) / 16;  // 512 threads = 16 waves/block

  // normalization
  k_init_deg <<<gN, 256, 0, stream>>>(dis, N);
  k_count_deg<<<gE, 256, 0, stream>>>(row, dis, E);
  k_dis      <<<gN, 256, 0, stream>>>(dis, N);
  k_norm     <<<gE, 256, 0, stream>>>(row, col, dis, normv, E);

  // layer 1
  k_gemm128  <<<gTil, 256, 0, stream>>>(X,  W1, b1, h0, N, 0);
  k_agg_self <<<gNH, 256, 0, stream>>>(h0, dis, h1, NH);
  k_agg_edges<<<gEdg, 512, 0, stream>>>(h0, row, col, normv, h1, E);
  // layer 2 (ReLU folded into GEMM input load)
  k_gemm128  <<<gTil, 256, 0, stream>>>(h1, W2, b2, h0, N, 1);
  k_agg_self <<<gNH, 256, 0, stream>>>(h0, dis, h1, NH);
  k_agg_edges<<<gEdg, 512, 0, stream>>>(h0, row, col, normv, h1, E);
  // layer 3
  k_gemm128  <<<gTil, 256, 0, stream>>>(h1, W3, b3, h0, N, 1);
  k_agg_self <<<gNH, 256, 0, stream>>>(h0, dis, h1, NH);
  k_agg_edges<<<gEdg, 512, 0, stream>>>(h0, row, col, normv, h1, E);

  // pooling (ReLU folded into accumulation)
  const unsigned gGH = (G * HID + 255) / 256;
  k_pool_zero <<<gGH, 256, 0, stream>>>(s, mx, cnt, G);
  k_pool_accum<<<gNH, 256, 0, stream>>>(h1, batch, s, (unsigned int*)mx, cnt, NH);
  k_pool_final<<<(G * 2 * HID + 255) / 256, 256, 0, stream>>>(s, mx, cnt, gbuf, G);

  // MLP head
  k_mlp1<<<(G * HID + 255) / 256, 256, 0, stream>>>(gbuf, Wm1, bm1, hm, G);
  k_mlp2<<<(G * 10 + 63) / 64, 64, 0, stream>>>(hm, Wm2, bm2, out, G);
}